// GATLayer_86852828659809
// MI455X (gfx1250) — compile-verified
//
#include <hip/hip_runtime.h>
#include <math.h>

#define N_NODES 50000
#define N_EDGES 800000
#define F_IN    128
#define HEADS   8
#define HID     16
#define HC      128        // HEADS*HID
#define EPS     1e-5f
#define GAT_SLOPE 0.2f
#define ACT_SLOPE 0.01f

typedef __attribute__((ext_vector_type(2))) float v2f;
typedef __attribute__((ext_vector_type(8))) float v8f;

// ---- order-preserving float <-> int key (for integer atomicMax on floats) ----
__device__ __forceinline__ int f2key(float f) {
    int b = __float_as_int(f);
    return b ^ ((b >> 31) & 0x7FFFFFFF);
}
__device__ __forceinline__ float key2f(int k) {
    int b = k ^ ((k >> 31) & 0x7FFFFFFF);
    return __int_as_float(b);
}

// ---------------------------------------------------------------------------
// 0) workspace / output init (graph replays don't re-poison: must re-init)
// ---------------------------------------------------------------------------
__global__ void init_ws(float* __restrict__ out, float* __restrict__ denom,
                        int* __restrict__ lmaxKey, float* __restrict__ sums) {
    int i = blockIdx.x * blockDim.x + threadIdx.x;
    if (i < N_NODES * HC) out[i] = 0.0f;
    if (i < N_NODES * HEADS) { denom[i] = 0.0f; lmaxKey[i] = (int)0x80000000; }
    if (i < 2 * HC) sums[i] = 0.0f;
}

// ---------------------------------------------------------------------------
// 1) xl = x@Wl + bl ; xr = x@Wr + br   via V_WMMA_F32_16X16X4_F32 (full fp32)
//    block = 16 rows of x, 8 waves; wave w owns output cols [16w,16w+16) of
//    BOTH Wl and Wr (A-tile reused across the two accumulators).
// ---------------------------------------------------------------------------
__global__ __launch_bounds__(256) void gemm_xlxr(
    const float* __restrict__ x,
    const float* __restrict__ Wl, const float* __restrict__ bl,
    const float* __restrict__ Wr, const float* __restrict__ br,
    float* __restrict__ xl, float* __restrict__ xr)
{
    const int lane    = threadIdx.x & 31;
    const int wave    = threadIdx.x >> 5;
    const int rowBase = blockIdx.x * 16;     // 3125 * 16 == 50000 exactly
    const int n0      = wave * 16;
    const int mrow    = lane & 15;
    const int hi      = lane >> 4;           // half-wave selects K pair
    const int col     = n0 + mrow;

    v8f accL = {};
    v8f accR = {};
    const float* xrow = x + (size_t)(rowBase + mrow) * F_IN + 2 * hi;

    #pragma unroll 4
    for (int k0 = 0; k0 < F_IN; k0 += 4) {
        // A 16x4 fp32: lanes0-15 -> K=k0,k0+1 ; lanes16-31 -> K=k0+2,k0+3
        v2f a = *(const v2f*)(xrow + k0);
        const int krow = k0 + 2 * hi;
        v2f bL, bR;
        bL.x = Wl[(size_t)krow * HC + col];
        bL.y = Wl[(size_t)(krow + 1) * HC + col];
        bR.x = Wr[(size_t)krow * HC + col];
        bR.y = Wr[(size_t)(krow + 1) * HC + col];
        accL = __builtin_amdgcn_wmma_f32_16x16x4_f32(false, a, false, bL,
                                                     (short)0, accL, false, false);
        accR = __builtin_amdgcn_wmma_f32_16x16x4_f32(false, a, false, bR,
                                                     (short)0, accR, false, false);
    }

    const float biasL = bl[col];
    const float biasR = br[col];
    #pragma unroll
    for (int v = 0; v < 8; ++v) {
        const int row = rowBase + v + 8 * hi;   // C/D layout: vgpr v -> M=v / v+8
        xl[(size_t)row * HC + col] = accL[v] + biasL;
        xr[(size_t)row * HC + col] = accR[v] + biasR;
    }
}

// ---------------------------------------------------------------------------
// 2) per-edge attention logits + segment max (2 edges/block, 128 thr/edge)
// ---------------------------------------------------------------------------
__global__ __launch_bounds__(256) void edge_logits(
    const float* __restrict__ xl, const float* __restrict__ xr,
    const int* __restrict__ eidx, const float* __restrict__ ea,
    const float* __restrict__ We, const float* __restrict__ be,
    const float* __restrict__ att,
    float* __restrict__ logits, int* __restrict__ lmaxKey)
{
    const int t = threadIdx.x;
    const int e = blockIdx.x * 2 + (t >> 7);   // grid = N_EDGES/2 (E even)
    const int c = t & 127;

    const int src = eidx[e];
    const int dst = eidx[N_EDGES + e];

    // fused edge-feature transform: e = edge_attr @ We + be (K=2, do inline)
    const float ev = ea[e * 2] * We[c] + ea[e * 2 + 1] * We[HC + c] + be[c];
    float m = xl[(size_t)src * HC + c] + xr[(size_t)dst * HC + c] + ev;
    m = m > 0.0f ? m : GAT_SLOPE * m;          // GATv2 leaky_relu(0.2)
    float p = m * att[c];                      // att flat [h*16 + j] == att[c]

    // reduce across the 16 lanes of this head (wave32 -> width-16 segments)
    p += __shfl_xor(p, 8, 16);
    p += __shfl_xor(p, 4, 16);
    p += __shfl_xor(p, 2, 16);
    p += __shfl_xor(p, 1, 16);

    if ((c & 15) == 0) {
        const int h = c >> 4;
        logits[(size_t)e * HEADS + h] = p;
        atomicMax(&lmaxKey[dst * HEADS + h], f2key(p));   // L2-resident atomic
    }
}

// ---------------------------------------------------------------------------
// 3) ex = exp(logit - lmax[dst]); denom[dst] += ex   (one thread per edge,head)
// ---------------------------------------------------------------------------
__global__ void edge_exp(const int* __restrict__ eidx, float* __restrict__ logits,
                         const int* __restrict__ lmaxKey, float* __restrict__ denom)
{
    const int idx = blockIdx.x * blockDim.x + threadIdx.x;
    if (idx >= N_EDGES * HEADS) return;
    const int e = idx >> 3;
    const int h = idx & 7;
    const int dst = eidx[N_EDGES + e];
    const float lmax = key2f(lmaxKey[dst * HEADS + h]);
    const float ex = expf(logits[idx] - lmax);
    logits[idx] = ex;                                  // overwrite in place
    atomicAdd(&denom[dst * HEADS + h], ex);
}

// ---------------------------------------------------------------------------
// 4) out[dst] += alpha * xl[src]   (2 edges/block, coalesced gather+scatter)
// ---------------------------------------------------------------------------
__global__ __launch_bounds__(256) void edge_agg(
    const float* __restrict__ xl, const int* __restrict__ eidx,
    const float* __restrict__ logits, const float* __restrict__ denom,
    float* __restrict__ out)
{
    const int t = threadIdx.x;
    const int e = blockIdx.x * 2 + (t >> 7);
    const int c = t & 127;
    const int h = c >> 4;
    const int src = eidx[e];
    const int dst = eidx[N_EDGES + e];
    const float alpha = logits[(size_t)e * HEADS + h] / denom[dst * HEADS + h];
    atomicAdd(&out[(size_t)dst * HC + c], xl[(size_t)src * HC + c] * alpha);
}

// ---------------------------------------------------------------------------
// 5) column sums/sumsq for BatchNorm (bias cancels inside BN -> skipped)
// ---------------------------------------------------------------------------
__global__ __launch_bounds__(128) void col_stats(const float* __restrict__ out,
                                                 float* __restrict__ sums)
{
    const int c  = threadIdx.x;                  // channel
    const int r0 = blockIdx.x * 128;
    const int r1 = (r0 + 128 < N_NODES) ? (r0 + 128) : N_NODES;
    float s = 0.0f, s2 = 0.0f;
    for (int r = r0; r < r1; ++r) {
        const float v = out[(size_t)r * HC + c];
        s += v; s2 += v * v;
    }
    atomicAdd(&sums[c], s);
    atomicAdd(&sums[HC + c], s2);
}

// ---------------------------------------------------------------------------
// 6) BatchNorm (batch stats) + LeakyReLU(0.01), in place on d_out
// ---------------------------------------------------------------------------
__global__ void bn_act(float* __restrict__ out, const float* __restrict__ sums,
                       const float* __restrict__ gamma, const float* __restrict__ beta)
{
    const int idx = blockIdx.x * blockDim.x + threadIdx.x;
    if (idx >= N_NODES * HC) return;
    const int c = idx & 127;
    const float invN = 1.0f / (float)N_NODES;
    const float mean = sums[c] * invN;
    const float var  = sums[HC + c] * invN - mean * mean;
    float v = (out[idx] - mean) * rsqrtf(var + EPS) * gamma[c] + beta[c];
    out[idx] = v > 0.0f ? v : ACT_SLOPE * v;
}

// ---------------------------------------------------------------------------
extern "C" void kernel_launch(void* const* d_in, const int* in_sizes, int n_in,
                              void* d_out, int out_size, void* d_ws, size_t ws_size,
                              hipStream_t stream)
{
    (void)in_sizes; (void)n_in; (void)out_size; (void)ws_size;

    const float* x    = (const float*)d_in[0];
    const int*   eidx = (const int*)  d_in[1];
    const float* ea   = (const float*)d_in[2];
    const float* Wl   = (const float*)d_in[3];
    const float* bl   = (const float*)d_in[4];
    const float* Wr   = (const float*)d_in[5];
    const float* br   = (const float*)d_in[6];
    const float* We   = (const float*)d_in[7];
    const float* be   = (const float*)d_in[8];
    const float* att  = (const float*)d_in[9];
    // d_in[10] = bias: cancels analytically inside BatchNorm -> unused
    const float* gamma = (const float*)d_in[11];
    const float* beta  = (const float*)d_in[12];

    float* out = (float*)d_out;

    // workspace layout (floats): xl | xr | logits | denom | lmaxKey | sums
    float* ws      = (float*)d_ws;
    float* xl      = ws;                                   // N*HC  = 6.4M
    float* xr      = xl + (size_t)N_NODES * HC;            // N*HC  = 6.4M
    float* logits  = xr + (size_t)N_NODES * HC;            // E*H   = 6.4M
    float* denom   = logits + (size_t)N_EDGES * HEADS;     // N*H   = 0.4M
    int*   lmaxKey = (int*)(denom + (size_t)N_NODES * HEADS);
    float* sums    = (float*)(lmaxKey + (size_t)N_NODES * HEADS); // 2*HC

    const int initGrid = (N_NODES * HC + 255) / 256;       // 25000
    init_ws<<<initGrid, 256, 0, stream>>>(out, denom, lmaxKey, sums);

    gemm_xlxr<<<N_NODES / 16, 256, 0, stream>>>(x, Wl, bl, Wr, br, xl, xr);

    edge_logits<<<N_EDGES / 2, 256, 0, stream>>>(xl, xr, eidx, ea, We, be, att,
                                                 logits, lmaxKey);

    edge_exp<<<(N_EDGES * HEADS + 255) / 256, 256, 0, stream>>>(eidx, logits,
                                                                lmaxKey, denom);

    edge_agg<<<N_EDGES / 2, 256, 0, stream>>>(xl, eidx, logits, denom, out);

    col_stats<<<(N_NODES + 127) / 128, 128, 0, stream>>>(out, sums);

    bn_act<<<initGrid, 256, 0, stream>>>(out, sums, gamma, beta);
}